// HypAveragedHausdorffLoss_76716705841702
// MI455X (gfx1250) — compile-verified
//
#include <hip/hip_runtime.h>
#include <math.h>

typedef __attribute__((ext_vector_type(2))) float v2f;
typedef __attribute__((ext_vector_type(8))) float v8f;

static constexpr int DIM = 16;          // point dimensionality
static constexpr int TS  = 16;          // WMMA tile size
static constexpr int WPB = 8;           // waves per block (256 threads)
static constexpr int CSPLIT = 8;        // column-space splits per row tile
static constexpr float FMAX = 3.402823466e+38f;

// ---------------------------------------------------------------------------
// Kernel 1: per-point info {-0.5*norm^2, 1/(1-norm^2)} + init min buffers.
// Storing -norm^2/2 lets the main kernel fold the (xn+yn) bias straight into
// the WMMA accumulator with a single pk_add.
// ---------------------------------------------------------------------------
__global__ void hyp_info_init(const float* __restrict__ X, const float* __restrict__ Y,
                              int n1, int n2,
                              float2* __restrict__ xinfo, float2* __restrict__ yinfo,
                              unsigned* __restrict__ rowmin, unsigned* __restrict__ colmin)
{
    int i = blockIdx.x * blockDim.x + threadIdx.x;
    if (i < n1) {
        const float* p = X + (size_t)i * DIM;
        float s = 0.f;
        #pragma unroll
        for (int k = 0; k < DIM; ++k) s = fmaf(p[k], p[k], s);
        xinfo[i] = make_float2(-0.5f * s, 1.0f / (1.0f - s));
        rowmin[i] = __float_as_uint(FMAX);
    } else if (i < n1 + n2) {
        int j = i - n1;
        const float* p = Y + (size_t)j * DIM;
        float s = 0.f;
        #pragma unroll
        for (int k = 0; k < DIM; ++k) s = fmaf(p[k], p[k], s);
        yinfo[j] = make_float2(-0.5f * s, 1.0f / (1.0f - s));
        colmin[j] = __float_as_uint(FMAX);
    }
}

// ---------------------------------------------------------------------------
// Kernel 2: row-min of s = sqdist*invx*invy via WMMA f32 16x16x4 (K=16 -> 4 ops).
//   C-init = -(xn+yn)/2  =>  acc = dot - (xn+yn)/2 = -sqdist/2
//   tracked p = acc * (-2*invy) = sqdist*invy ; invx applied once at the end.
// outmin: float bits, uint atomicMin (s >= 0 after clamp).
// ---------------------------------------------------------------------------
__global__ void hyp_min_pass(const float*  __restrict__ P,
                             const float2* __restrict__ pinfo,
                             const float*  __restrict__ Q,
                             const float2* __restrict__ qinfo,
                             unsigned* __restrict__ outmin,
                             int nrows, int ncols)
{
    const int lane  = threadIdx.x & 31;
    const int wave  = threadIdx.x >> 5;
    const int gwave = blockIdx.x * WPB + wave;

    const int ntr = nrows / TS;          // row tiles
    if (gwave >= ntr * CSPLIT) return;   // wave-uniform guard (EXEC stays all-1s)

    const int rt    = gwave % ntr;
    const int chunk = gwave / ntr;
    const int ctiles   = ncols / TS;
    const int ct_begin = (ctiles * chunk) / CSPLIT;
    const int ct_end   = (ctiles * (chunk + 1)) / CSPLIT;

    const int hi = lane >> 4;            // half-wave select (K/M high part)
    const int lp = lane & 15;
    const int rowbase = rt * TS;

    // A fragments: 16x4 f32, lane layout per ISA 7.12.2:
    //   v.x = P[row][4f + 2*hi], v.y = P[row][4f + 2*hi + 1], row = lane%16
    const float* prow = P + (size_t)(rowbase + lp) * DIM + 2 * hi;
    v2f a0 = *(const v2f*)(prow + 0);
    v2f a1 = *(const v2f*)(prow + 4);
    v2f a2 = *(const v2f*)(prow + 8);
    v2f a3 = *(const v2f*)(prow + 12);

    // Per-lane info for the 8 C/D rows this lane owns (M = v + 8*hi)
    float nxn[8], xi[8];                 // nxn = -xn/2
    #pragma unroll
    for (int v = 0; v < 8; ++v) {
        float2 t = pinfo[rowbase + v + 8 * hi];
        nxn[v] = t.x; xi[v] = t.y;
    }

    float rmin[8];
    #pragma unroll
    for (int v = 0; v < 8; ++v) rmin[v] = FMAX;

    // Pointer-bumped column-tile stream (B fragments share A's float2 pattern)
    const float*  qrow = Q + (size_t)(ct_begin * TS + lp) * DIM + 2 * hi;
    const float2* qip  = qinfo + (size_t)(ct_begin * TS + lp);

    for (int ct = ct_begin; ct < ct_end; ++ct, qrow += TS * DIM, qip += TS) {
        v2f b0 = *(const v2f*)(qrow + 0);
        v2f b1 = *(const v2f*)(qrow + 4);
        v2f b2 = *(const v2f*)(qrow + 8);
        v2f b3 = *(const v2f*)(qrow + 12);
        float2 qi = *qip;

        // Speculative prefetch of next tile (invalid addr silently dropped)
        __builtin_prefetch(qrow + TS * DIM, 0, 0);

        // Bias accumulator: C[v] = -(xn_v + yn)/2  (single pk_add chain)
        const float qy = qi.x;           // -yn/2
        v8f acc;
        #pragma unroll
        for (int v = 0; v < 8; ++v) acc[v] = nxn[v] + qy;

        acc = __builtin_amdgcn_wmma_f32_16x16x4_f32(false, a0, false, b0, (short)0, acc, false, false);
        acc = __builtin_amdgcn_wmma_f32_16x16x4_f32(false, a1, false, b1, (short)0, acc, false, false);
        acc = __builtin_amdgcn_wmma_f32_16x16x4_f32(false, a2, false, b2, (short)0, acc, false, false);
        acc = __builtin_amdgcn_wmma_f32_16x16x4_f32(false, a3, false, b3, (short)0, acc, false, false);

        // acc = -sqdist/2  ->  p = sqdist*invy
        const float z = -2.0f * qi.y;
        #pragma unroll
        for (int v = 0; v < 8; ++v)
            rmin[v] = fminf(rmin[v], acc[v] * z);
    }

    // Cross-lane min within each 16-lane half (rows 0..7 in lo, 8..15 in hi)
    #pragma unroll
    for (int m = 8; m >= 1; m >>= 1) {
        #pragma unroll
        for (int v = 0; v < 8; ++v)
            rmin[v] = fminf(rmin[v], __shfl_xor(rmin[v], m, 32));
    }
    if (lp == 0) {
        #pragma unroll
        for (int v = 0; v < 8; ++v) {
            float s = fmaxf(rmin[v] * xi[v], 0.0f);      // apply invx, clamp rounding
            atomicMin(&outmin[rowbase + v + 8 * hi], __float_as_uint(s));
        }
    }
}

// ---------------------------------------------------------------------------
// Kernel 3: arcosh(1 + 2s) over the N+M mins, two means, scalar output.
// ---------------------------------------------------------------------------
__global__ void hyp_finalize(const unsigned* __restrict__ rowmin,
                             const unsigned* __restrict__ colmin,
                             int n1, int n2, float* __restrict__ out)
{
    __shared__ float sh[256];
    const float w1 = 1.0f / (float)n1, w2 = 1.0f / (float)n2;
    float acc = 0.f;
    for (int i = threadIdx.x; i < n1; i += 256) {
        float s = __uint_as_float(rowmin[i]);
        float u = fmaf(2.f, s, 1.f);
        float r = sqrtf(fmaxf(fmaf(u, u, -1.f), 0.f));
        acc = fmaf(logf(u + r), w1, acc);
    }
    for (int i = threadIdx.x; i < n2; i += 256) {
        float s = __uint_as_float(colmin[i]);
        float u = fmaf(2.f, s, 1.f);
        float r = sqrtf(fmaxf(fmaf(u, u, -1.f), 0.f));
        acc = fmaf(logf(u + r), w2, acc);
    }
    sh[threadIdx.x] = acc;
    __syncthreads();
    for (int off = 128; off > 0; off >>= 1) {
        if ((int)threadIdx.x < off) sh[threadIdx.x] += sh[threadIdx.x + off];
        __syncthreads();
    }
    if (threadIdx.x == 0) out[0] = sh[0];
}

// ---------------------------------------------------------------------------
extern "C" void kernel_launch(void* const* d_in, const int* in_sizes, int n_in,
                              void* d_out, int out_size, void* d_ws, size_t ws_size,
                              hipStream_t stream) {
    const float* X = (const float*)d_in[0];
    const float* Y = (const float*)d_in[1];
    const int n1 = in_sizes[0] / DIM;
    const int n2 = in_sizes[1] / DIM;

    // Workspace layout (all overwritten before use; ~400 KB total)
    float2*   xinfo  = (float2*)d_ws;                       // n1
    float2*   yinfo  = xinfo + n1;                          // n2
    unsigned* rowmin = (unsigned*)(yinfo + n2);             // n1
    unsigned* colmin = rowmin + n1;                         // n2

    const int total = n1 + n2;
    hyp_info_init<<<(total + 255) / 256, 256, 0, stream>>>(
        X, Y, n1, n2, xinfo, yinfo, rowmin, colmin);

    const int waves1 = (n1 / TS) * CSPLIT;
    hyp_min_pass<<<(waves1 + WPB - 1) / WPB, WPB * 32, 0, stream>>>(
        X, xinfo, Y, yinfo, rowmin, n1, n2);

    const int waves2 = (n2 / TS) * CSPLIT;
    hyp_min_pass<<<(waves2 + WPB - 1) / WPB, WPB * 32, 0, stream>>>(
        Y, yinfo, X, xinfo, colmin, n2, n1);

    hyp_finalize<<<1, 256, 0, stream>>>(rowmin, colmin, n1, n2, (float*)d_out);
}